// BinaryPathEncoder_74646531604770
// MI455X (gfx1250) — compile-verified
//
#include <hip/hip_runtime.h>

typedef __attribute__((ext_vector_type(2))) float v2f;
typedef __attribute__((ext_vector_type(4))) float v4f;
typedef __attribute__((ext_vector_type(8))) float v8f;

#define DIM 128
#define MAT_ELEMS (DIM * DIM)      // 16384 floats (global row-major / swizzled-B image)
#define LDB 132                    // A/D-side LDS row stride: 132 ≡ 4 (mod 64 banks) -> conflict-free A loads & D stores
#define MATP (DIM * LDB)           // 16896 floats per padded LDS matrix
#define NSTEPS 16

// ---------------------------------------------------------------------------
// Swizzled B layout: for k-chunk kc = k>>2 and j-tile jt = j>>4, fragment index
// f = kc*8 + jt. Lane l = (k>>1 & 1)*16 + (j & 15) holds the v2f
// { B[4kc+2*half][j], B[4kc+2*half+1][j] } at Bs[f*64 + l*2].
// One ds_load_b64 per lane per fragment; 32 lanes cover all 64 banks once.
// ---------------------------------------------------------------------------
__device__ __forceinline__ int bswz(int k, int j)
{
    return ((k >> 2) * 8 + (j >> 4)) * 64 + ((k >> 1) & 1) * 32 + (j & 15) * 2 + (k & 1);
}

// Out(strip) = A(strip) @ B.  A/Out: LDS, row stride LDB.  Bs: LDS, swizzled.
// Wave w owns output rows [16w, 16w+16) and reads only those rows of A, so
// in-place (Out == A) is safe per-wave without barriers (LDS in-order per wave).
__device__ __forceinline__ void strip_matmul(const float* A, const float* Bs, float* Out, int tid)
{
    const int lane = tid & 31;
    const int lo   = lane & 15;
    const int half = lane >> 4;
    const int m0   = (tid >> 5) * 16;

    v8f acc[8] = {};

#pragma unroll 2
    for (int k0 = 0; k0 < DIM; k0 += 4) {
        // A-frag 16x4: single b64 per lane, banks tiled perfectly (LDB % 64 == 4)
        v2f a = *(const v2f*)(A + (m0 + lo) * LDB + k0 + 2 * half);
        // preload all 8 B-frags of this K-chunk, then fire 8 WMMAs
        const float* bp = Bs + (k0 << 7) + (lane << 1);
        v2f bf[8];
#pragma unroll
        for (int jt = 0; jt < 8; ++jt)
            bf[jt] = *(const v2f*)(bp + (jt << 6));
#pragma unroll
        for (int jt = 0; jt < 8; ++jt)
            acc[jt] = __builtin_amdgcn_wmma_f32_16x16x4_f32(
                false, a, false, bf[jt], (short)0, acc[jt], false, false);
    }
#pragma unroll
    for (int jt = 0; jt < 8; ++jt) {
#pragma unroll
        for (int v = 0; v < 8; ++v) {
            // D layout: VGPR v -> row m0+v+8*half, col jt*16+lo; halves 32 banks apart
            Out[(m0 + v + 8 * half) * LDB + jt * 16 + lo] = acc[jt][v];
        }
    }
}

// Re-pack an LDB-layout matrix into swizzled-B order (block-wide, barrier outside).
__device__ __forceinline__ void swizzleB(const float* M, float* Bs, int tid)
{
    for (int e = tid; e < MAT_ELEMS; e += 256) {
        int k = e >> 7, j = e & 127;
        Bs[bswz(k, j)] = M[k * LDB + j];
    }
}

// ---------------------------------------------------------------------------
// expm of herm_k = P_k - P_k^T via scaling (2^-8) + 16-term Taylor (Horner)
// + 8 squarings; stores trans[k] = exp(herm_k)^T to ws (row-major 128).
// ---------------------------------------------------------------------------
__global__ __launch_bounds__(256) void expm_trans_kernel(const float* __restrict__ P,
                                                         float* __restrict__ transOut)
{
    extern __shared__ float smem[];
    float* Am = smem;                 // scaled A   (LDB layout)
    float* R  = smem + MATP;          // running    (LDB layout)
    float* T  = smem + 2 * MATP;      // temp       (LDB layout)
    float* Bs = smem + 3 * MATP;      // swizzled B (16384)
    const int tid = threadIdx.x;
    const float* Pk = P + (size_t)blockIdx.x * MAT_ELEMS;

    const float scale = 1.0f / 256.0f;  // 2^-8
    for (int e = tid; e < MAT_ELEMS; e += 256) {
        int i = e >> 7, j = e & 127;
        Am[i * LDB + j] = (Pk[i * DIM + j] - Pk[j * DIM + i]) * scale;
        R[i * LDB + j]  = (i == j) ? 1.0f : 0.0f;
    }
    __syncthreads();

    // Horner: for it = 16..1 : R = I + (Am @ R) / it
    for (int it = 16; it >= 1; --it) {
        swizzleB(R, Bs, tid);
        __syncthreads();
        strip_matmul(Am, Bs, T, tid);
        __syncthreads();
        const float inv = 1.0f / (float)it;
        for (int e = tid; e < MAT_ELEMS; e += 256) {
            int i = e >> 7, j = e & 127;
            R[i * LDB + j] = ((i == j) ? 1.0f : 0.0f) + T[i * LDB + j] * inv;
        }
        __syncthreads();
    }
    // 8 squarings: R = R @ R
    for (int sq = 0; sq < 8; ++sq) {
        swizzleB(R, Bs, tid);
        __syncthreads();
        strip_matmul(R, Bs, T, tid);
        __syncthreads();
        for (int e = tid; e < MAT_ELEMS; e += 256) {
            int i = e >> 7, j = e & 127;
            R[i * LDB + j] = T[i * LDB + j];
        }
        __syncthreads();
    }
    // trans[k][i][j] = exp(herm)[j][i]
    float* outk = transOut + (size_t)blockIdx.x * MAT_ELEMS;
    for (int e = tid; e < MAT_ELEMS; e += 256) {
        int i = e >> 7, j = e & 127;
        outk[e] = R[j * LDB + i];
    }
}

// ---------------------------------------------------------------------------
// One block per position: cur = prod_s trans[bit_s(pos)], early-exit when the
// path is exhausted (pos<2 -> identity).
// ---------------------------------------------------------------------------
__global__ __launch_bounds__(256) void path_scan_kernel(const int* __restrict__ unique,
                                                        const float* __restrict__ trans,
                                                        float* __restrict__ out)
{
    extern __shared__ float smem[];
    float* Bs  = smem;                     // trans[0], trans[1] swizzled (2*16384)
    float* cur = smem + 2 * MAT_ELEMS;     // LDB layout (16896)
    const int tid  = threadIdx.x;
    const int wave = tid >> 5;
    const int lane = tid & 31;
    const int b    = blockIdx.x;

    // Load trans (global row-major) directly into swizzled-B LDS images.
    const v4f* g4 = (const v4f*)trans;
    for (int f = tid; f < 2 * 4096; f += 256) {
        v4f v = g4[f];
        int e   = f << 2;
        int mat = e >> 14;
        int k   = (e >> 7) & 127;
        int j   = e & 127;                       // multiple of 4, same j-tile
        float* dst = Bs + mat * MAT_ELEMS + bswz(k, j);
        dst[0] = v.x; dst[2] = v.y; dst[4] = v.z; dst[6] = v.w;
    }
    // Identity init of this wave's strip of cur (wave-local, no barrier needed).
    for (int e = lane; e < 16 * DIM; e += 32) {
        int r = wave * 16 + (e >> 7);
        int c = e & 127;
        cur[r * LDB + c] = (r == c) ? 1.0f : 0.0f;
    }
    unsigned pos = (unsigned)unique[b];
    __syncthreads();   // swizzled trans visible to all waves

    // Sequential path composition; uniform control flow, waves run barrier-free.
    for (int s = 0; s < NSTEPS; ++s) {
        unsigned sh = pos >> s;
        if (sh < 2u) break;
        const float* Tm = Bs + (sh & 1u) * MAT_ELEMS;
        strip_matmul(cur, Tm, cur, tid);     // in-place, wave-local strip
    }
    __syncthreads();   // all strips complete before block-wide copy-out

    // Streaming (nontemporal) copy-out: 64 MiB write-once.
    v4f* o = (v4f*)(out + (size_t)b * MAT_ELEMS);
    for (int f = tid; f < 4096; f += 256) {
        int r = f >> 5, c4 = f & 31;
        v4f v = *(const v4f*)&cur[r * LDB + c4 * 4];
        __builtin_nontemporal_store(v, &o[f]);
    }
}

extern "C" void kernel_launch(void* const* d_in, const int* in_sizes, int n_in,
                              void* d_out, int out_size, void* d_ws, size_t ws_size,
                              hipStream_t stream)
{
    const int*   unique = (const int*)d_in[0];    // N positions
    const float* prims  = (const float*)d_in[1];  // [2,128,128] f32
    float*       out    = (float*)d_out;          // [N,128,128] f32
    float*       trans  = (float*)d_ws;           // 2*16384 f32 workspace

    const int n = in_sizes[0];
    const size_t lds_expm = (size_t)(3 * MATP + MAT_ELEMS) * sizeof(float);   // ~262 KB
    const size_t lds_scan = (size_t)(2 * MAT_ELEMS + MATP) * sizeof(float);   // ~194 KB

    expm_trans_kernel<<<dim3(2), dim3(256), lds_expm, stream>>>(prims, trans);
    path_scan_kernel<<<dim3(n), dim3(256), lds_scan, stream>>>(unique, trans, out);
}